// Model_39676907884714
// MI455X (gfx1250) — compile-verified
//
#include <hip/hip_runtime.h>
#include <hip/hip_bf16.h>

typedef __attribute__((ext_vector_type(2))) float v2f;
typedef __attribute__((ext_vector_type(8))) float v8f;

constexpr int M_ROWS      = 4096;   // query rows (x1)
constexpr int N_COLS      = 4096;   // key rows  (x2) == softmax axis
constexpr int D_DIM       = 1024;   // feature dim
constexpr int ROWS_PER_WG = 16;     // one 16-row score panel per workgroup
constexpr int SPITCH      = 4100;   // LDS row pitch in floats (8B-aligned rows, de-aliased banks)
constexpr int WAVES       = 8;      // 256 threads / wave32

// ---------------- JAX threefry2x32 (exact, 20 rounds) ----------------
__device__ __forceinline__ unsigned rotl32(unsigned x, int r) {
    return (x << r) | (x >> (32 - r));
}

__device__ __forceinline__ void threefry2x32_0_42(unsigned& x0, unsigned& x1) {
    const unsigned k0 = 0u, k1 = 42u;
    const unsigned ks2 = k0 ^ k1 ^ 0x1BD11BDAu;
    x0 += k0; x1 += k1;
#define TF_R(r) { x0 += x1; x1 = rotl32(x1, r); x1 ^= x0; }
    TF_R(13) TF_R(15) TF_R(26) TF_R(6)
    x0 += k1;  x1 += ks2 + 1u;
    TF_R(17) TF_R(29) TF_R(16) TF_R(24)
    x0 += ks2; x1 += k0 + 2u;
    TF_R(13) TF_R(15) TF_R(26) TF_R(6)
    x0 += k0;  x1 += k1 + 3u;
    TF_R(17) TF_R(29) TF_R(16) TF_R(24)
    x0 += k1;  x1 += ks2 + 4u;
    TF_R(13) TF_R(15) TF_R(26) TF_R(6)
    x0 += ks2; x1 += k0 + 5u;
#undef TF_R
}

// keep mask == jax.random.bernoulli(key(42), 0.5, (4096,4096))[row][col]
// iota(2^24) split into halves: pair (j, j+2^23); keep <=> uniform < 0.5 <=> MSB(bits)==0
__device__ __forceinline__ bool dropout_keep(unsigned row, unsigned col) {
    const unsigned which = (row >= 2048u) ? 1u : 0u;
    const unsigned j = (row - (which << 11)) * 4096u + col;
    unsigned x0 = j, x1 = j + 8388608u;   // j + 2^23
    threefry2x32_0_42(x0, x1);
    const unsigned bits = which ? x1 : x0;
    return (bits >> 31) == 0u;
}

// ---------------- wave32 reductions ----------------
__device__ __forceinline__ float wave_max32(float v) {
    #pragma unroll
    for (int off = 16; off > 0; off >>= 1)
        v = fmaxf(v, __shfl_xor(v, off, 32));
    return v;
}
__device__ __forceinline__ float wave_sum32(float v) {
    #pragma unroll
    for (int off = 16; off > 0; off >>= 1)
        v += __shfl_xor(v, off, 32);
    return v;
}

__device__ __forceinline__ v8f wmma_f32(v2f a, v2f b, v8f c) {
    return __builtin_amdgcn_wmma_f32_16x16x4_f32(
               false, a, false, b, (short)0, c, false, false);
}

// ---------------- fused attention kernel ----------------
__global__ void __launch_bounds__(256, 1)
attn_fused_kernel(const float* __restrict__ x1,
                  const float* __restrict__ x2,
                  float* __restrict__ out) {
    extern __shared__ float smem[];                 // ROWS_PER_WG x SPITCH floats
    const int row_blk = blockIdx.x * ROWS_PER_WG;
    const int tid     = threadIdx.x;
    const int wave    = tid >> 5;
    const int lane    = tid & 31;
    const int lane_lo = lane & 15;                  // M (A) / N (B,C) index within tile
    const int lane_hi = lane >> 4;                  // K-half selector

    // ================= GEMM1: S = 4 * x1 @ x2^T =================
    // 4 independent 16x16 C tiles per wave share one A fragment: 4 WMMA
    // accumulator chains in flight -> no WMMA->WMMA RAW stalls, A loads /4.
    const float2* x1r = (const float2*)(x1 + (size_t)(row_blk + lane_lo) * D_DIM);
    for (int grp = wave; grp < N_COLS / 64; grp += WAVES) {
        const int col_blk = grp * 64;
        const float2* x2r0 = (const float2*)(x2 + (size_t)(col_blk +  0 + lane_lo) * D_DIM);
        const float2* x2r1 = (const float2*)(x2 + (size_t)(col_blk + 16 + lane_lo) * D_DIM);
        const float2* x2r2 = (const float2*)(x2 + (size_t)(col_blk + 32 + lane_lo) * D_DIM);
        const float2* x2r3 = (const float2*)(x2 + (size_t)(col_blk + 48 + lane_lo) * D_DIM);
        v8f acc0 = {0.f,0.f,0.f,0.f,0.f,0.f,0.f,0.f};
        v8f acc1 = acc0, acc2 = acc0, acc3 = acc0;
        #pragma unroll 2
        for (int k0 = 0; k0 < D_DIM; k0 += 4) {
            const int kh = (k0 >> 1) + lane_hi;     // float2 index of (k0 + 2*lane_hi)
            const float2 a2 = x1r[kh];
            const float2 b0 = x2r0[kh];
            const float2 b1 = x2r1[kh];
            const float2 b2 = x2r2[kh];
            const float2 b3 = x2r3[kh];
            v2f av; av.x = a2.x; av.y = a2.y;
            v2f bv0; bv0.x = b0.x; bv0.y = b0.y;
            v2f bv1; bv1.x = b1.x; bv1.y = b1.y;
            v2f bv2; bv2.x = b2.x; bv2.y = b2.y;
            v2f bv3; bv3.x = b3.x; bv3.y = b3.y;
            acc0 = wmma_f32(av, bv0, acc0);
            acc1 = wmma_f32(av, bv1, acc1);
            acc2 = wmma_f32(av, bv2, acc2);
            acc3 = wmma_f32(av, bv3, acc3);
        }
        #pragma unroll
        for (int g = 0; g < 8; ++g) {               // C layout: VGPR g -> M = g + 8*lane_hi
            float* dst = smem + (g + 8 * lane_hi) * SPITCH + col_blk + lane_lo;
            dst[ 0] = acc0[g] * 4.0f;
            dst[16] = acc1[g] * 4.0f;
            dst[32] = acc2[g] * 4.0f;
            dst[48] = acc3[g] * 4.0f;
        }
    }
    __syncthreads();

    // ================= softmax + exact JAX dropout (rows in LDS) =================
    #pragma unroll
    for (int rr = 0; rr < 2; ++rr) {
        const int r = wave + rr * WAVES;            // 8 waves x 2 = 16 rows
        float* row = smem + r * SPITCH;
        float m = -3.402823466e+38f;
        for (int c = lane; c < N_COLS; c += 32) m = fmaxf(m, row[c]);
        m = wave_max32(m);
        float s = 0.f;
        for (int c = lane; c < N_COLS; c += 32) {
            const float e = __expf(row[c] - m);
            row[c] = e;
            s += e;
        }
        s = wave_sum32(s);
        const float f = 2.0f / s;                   // 1/sum * 1/(1-p)
        const unsigned grow = (unsigned)(row_blk + r);
        for (int c = lane; c < N_COLS; c += 32)
            row[c] = dropout_keep(grow, (unsigned)c) ? row[c] * f : 0.0f;
    }
    __syncthreads();

    // ================= GEMM2: out = P (LDS, 16x4096) @ x2 (4096x1024) =================
    // Same 4-tile register blocking: one ds_load_b64 A fragment feeds 4 WMMAs.
    const float* prow = smem + (size_t)lane_lo * SPITCH;
    for (int grp = wave; grp < D_DIM / 64; grp += WAVES) {   // 16 groups -> 2 per wave
        const int out_col = grp * 64;
        v8f acc0 = {0.f,0.f,0.f,0.f,0.f,0.f,0.f,0.f};
        v8f acc1 = acc0, acc2 = acc0, acc3 = acc0;
        #pragma unroll 2
        for (int k0 = 0; k0 < N_COLS; k0 += 4) {
            const int ka = k0 + 2 * lane_hi;
            const float2 a2 = *(const float2*)(prow + ka);   // ds_load_b64
            v2f av; av.x = a2.x; av.y = a2.y;
            const float* b0p = x2 + (size_t)ka * D_DIM + out_col + lane_lo;
            const float* b1p = b0p + D_DIM;                  // row ka+1
            v2f bv0; bv0.x = b0p[ 0]; bv0.y = b1p[ 0];
            v2f bv1; bv1.x = b0p[16]; bv1.y = b1p[16];
            v2f bv2; bv2.x = b0p[32]; bv2.y = b1p[32];
            v2f bv3; bv3.x = b0p[48]; bv3.y = b1p[48];
            acc0 = wmma_f32(av, bv0, acc0);
            acc1 = wmma_f32(av, bv1, acc1);
            acc2 = wmma_f32(av, bv2, acc2);
            acc3 = wmma_f32(av, bv3, acc3);
        }
        #pragma unroll
        for (int g = 0; g < 8; ++g) {
            float* dst = out + (size_t)(row_blk + g + 8 * lane_hi) * D_DIM + out_col + lane_lo;
            dst[ 0] = acc0[g];
            dst[16] = acc1[g];
            dst[32] = acc2[g];
            dst[48] = acc3[g];
        }
    }
}

extern "C" void kernel_launch(void* const* d_in, const int* in_sizes, int n_in,
                              void* d_out, int out_size, void* d_ws, size_t ws_size,
                              hipStream_t stream) {
    (void)in_sizes; (void)n_in; (void)d_ws; (void)ws_size; (void)out_size;
    const float* x1 = (const float*)d_in[0];
    const float* x2 = (const float*)d_in[1];
    float* out = (float*)d_out;

    const dim3 grid(M_ROWS / ROWS_PER_WG);          // 256 workgroups
    const dim3 block(256);                          // 8 wave32s
    const size_t shmem = (size_t)ROWS_PER_WG * SPITCH * sizeof(float); // 262,400 B < 320 KB
    attn_fused_kernel<<<grid, block, shmem, stream>>>(x1, x2, out);
}